// SupervisedNegLoss_62113817035255
// MI455X (gfx1250) — compile-verified
//
#include <hip/hip_runtime.h>
#include <hip/hip_bf16.h>
#include <math.h>

typedef __attribute__((ext_vector_type(2))) float v2f;
typedef __attribute__((ext_vector_type(4))) float v4f;
typedef __attribute__((ext_vector_type(8))) float v8f;

#define B_ROWS 4096
#define M_ROWS 8192
#define DIM    128
#define NTILE_BB  (B_ROWS / 16)               // 256 in-batch column tiles
#define NTILE_TOT ((B_ROWS + M_ROWS) / 16)    // 768 total column tiles
#define WAVES_PER_BLOCK 8

// ---------------------------------------------------------------------------
// Kernel 1: row L2-normalization. One wave (32 lanes) per row; each lane owns
// a float4 (32*4 = 128 = DIM). Butterfly shuffle reduce for sum of squares.
// ---------------------------------------------------------------------------
__global__ __launch_bounds__(256) void normalize_rows(
    const float* __restrict__ f, const float* __restrict__ fneg,
    float* __restrict__ outF, float* __restrict__ outG)
{
    const int wave = threadIdx.x >> 5;
    const int lane = threadIdx.x & 31;
    const int row  = blockIdx.x * WAVES_PER_BLOCK + wave;

    const float* src;
    float* dst;
    if (row < B_ROWS) {
        src = f    + (size_t)row * DIM;
        dst = outF + (size_t)row * DIM;
    } else {
        const int r = row - B_ROWS;
        src = fneg + (size_t)r * DIM;
        dst = outG + (size_t)r * DIM;
    }

    v4f x = *(const v4f*)(src + lane * 4);
    float ss = x.x * x.x + x.y * x.y + x.z * x.z + x.w * x.w;
    #pragma unroll
    for (int off = 1; off < 32; off <<= 1)
        ss += __shfl_xor(ss, off, 32);

    const float inv = 1.0f / fmaxf(sqrtf(ss), 1e-8f);
    x.x *= inv; x.y *= inv; x.z *= inv; x.w *= inv;
    *(v4f*)(dst + lane * 4) = x;
}

// ---------------------------------------------------------------------------
// Kernel 2: fused sim + exp-sum + positive-mask accumulation.
// One block per 16-row tile of fn. Each of the 8 waves keeps the 16x128
// A-fragment resident in 64 VGPRs and sweeps a strided subset of the 768
// 16-column tiles (256 in-batch tiles from fn, 512 from the negative bank).
// K = 128 -> 32 v_wmma_f32_16x16x4_f32 per tile, split across TWO
// independent accumulator chains (even/odd k-chunks) so the matrix pipe is
// not serialized on a single D->C dependency chain.
//
// Fragment layout (ISA 7.12.2, f32):
//   A (16x4, MxK):  lane = 16*h + m -> A.v[r] = A[m][2h + r]
//   B (4x16, KxN):  lane = 16*h + m -> B.v[r] = B[2h + r][m]  (== row-major
//                   row m of the second matrix => same load pattern as A)
//   C/D (16x16):    lane = 16*h + m, VGPR r -> D[r + 8h][m]
// ---------------------------------------------------------------------------
__global__ __launch_bounds__(256) void supcon_kernel(
    const float* __restrict__ fn, const float* __restrict__ gn,
    const int* __restrict__ labels, float* __restrict__ lossOut)
{
    __shared__ float sDen[WAVES_PER_BLOCK][16];
    __shared__ float sPos[WAVES_PER_BLOCK][16];
    __shared__ float sNp [WAVES_PER_BLOCK][16];

    const int w    = threadIdx.x >> 5;
    const int lane = threadIdx.x & 31;
    const int h    = lane >> 4;        // half: 0 -> lanes 0-15, 1 -> 16-31
    const int m    = lane & 15;
    const int i0   = blockIdx.x * 16;  // first row of this block's tile

    // A fragments: rows i0..i0+15, all 32 K-chunks, resident in registers.
    v2f a[32];
    const float* arow = fn + (size_t)(i0 + m) * DIM + 2 * h;
    #pragma unroll
    for (int c = 0; c < 32; ++c)
        a[c] = *(const v2f*)(arow + 4 * c);

    // Labels of the 8 rows this lane's accumulator VGPRs map to (r + 8h).
    int li[8];
    #pragma unroll
    for (int r = 0; r < 8; ++r)
        li[r] = labels[i0 + r + 8 * h];

    float dacc[8], pacc[8], ncnt[8];
    #pragma unroll
    for (int r = 0; r < 8; ++r) { dacc[r] = 0.f; pacc[r] = 0.f; ncnt[r] = 0.f; }

    for (int jt = w; jt < NTILE_TOT; jt += WAVES_PER_BLOCK) {
        const bool bb = (jt < NTILE_BB);
        const float* gsrc = bb ? (fn + (size_t)(jt * 16) * DIM)
                               : (gn + (size_t)((jt - NTILE_BB) * 16) * DIM);
        const float* brow = gsrc + (size_t)m * DIM + 2 * h;

        // Two independent WMMA accumulation chains (even / odd k-chunks).
        v8f acc0 = {};
        v8f acc1 = {};
        #pragma unroll
        for (int c = 0; c < 32; c += 2) {
            v2f b0 = *(const v2f*)(brow + 4 * c);
            v2f b1 = *(const v2f*)(brow + 4 * (c + 1));
            acc0 = __builtin_amdgcn_wmma_f32_16x16x4_f32(
                false, a[c],     false, b0, (short)0, acc0, false, false);
            acc1 = __builtin_amdgcn_wmma_f32_16x16x4_f32(
                false, a[c + 1], false, b1, (short)0, acc1, false, false);
        }

        if (bb) {
            const int j  = jt * 16 + m;
            const int lj = labels[j];
            #pragma unroll
            for (int r = 0; r < 8; ++r) {
                const int   i    = i0 + r + 8 * h;
                const float s    = acc0[r] + acc1[r];
                const bool  self = (j == i);
                dacc[r] += self ? 0.f : __expf(s);
                const bool pos = (lj == li[r]) && !self;
                pacc[r] += pos ? s : 0.f;
                ncnt[r] += pos ? 1.f : 0.f;
            }
        } else {
            #pragma unroll
            for (int r = 0; r < 8; ++r)
                dacc[r] += __expf(acc0[r] + acc1[r]);
        }
    }

    // Reduce over the column (m) dimension: butterfly within each 16-lane half.
    #pragma unroll
    for (int r = 0; r < 8; ++r) {
        #pragma unroll
        for (int off = 1; off < 16; off <<= 1) {
            dacc[r] += __shfl_xor(dacc[r], off, 32);
            pacc[r] += __shfl_xor(pacc[r], off, 32);
            ncnt[r] += __shfl_xor(ncnt[r], off, 32);
        }
    }

    if (m == 0) {
        #pragma unroll
        for (int r = 0; r < 8; ++r) {
            sDen[w][r + 8 * h] = dacc[r];
            sPos[w][r + 8 * h] = pacc[r];
            sNp [w][r + 8 * h] = ncnt[r];
        }
    }
    __syncthreads();

    // Fixed-order cross-wave combine (deterministic; no atomics).
    if (threadIdx.x < 16) {
        float den = 0.f, pos = 0.f, np = 0.f;
        #pragma unroll
        for (int v = 0; v < WAVES_PER_BLOCK; ++v) {
            den += sDen[v][threadIdx.x];
            pos += sPos[v][threadIdx.x];
            np  += sNp [v][threadIdx.x];
        }
        const float loss = (np > 0.f) ? (__logf(den) - pos / np) : 0.f;
        lossOut[i0 + threadIdx.x] = loss;
    }
}

// ---------------------------------------------------------------------------
// Kernel 3: deterministic fixed-order reduction of 4096 per-row losses.
// ---------------------------------------------------------------------------
__global__ __launch_bounds__(256) void reduce_loss(
    const float* __restrict__ lossIn, float* __restrict__ out)
{
    __shared__ float s[256];
    float acc = 0.f;
    for (int k = threadIdx.x; k < B_ROWS; k += 256)
        acc += lossIn[k];
    s[threadIdx.x] = acc;
    __syncthreads();
    #pragma unroll
    for (int off = 128; off > 0; off >>= 1) {
        if (threadIdx.x < off) s[threadIdx.x] += s[threadIdx.x + off];
        __syncthreads();
    }
    if (threadIdx.x == 0) out[0] = s[0] / (float)B_ROWS;
}

extern "C" void kernel_launch(void* const* d_in, const int* in_sizes, int n_in,
                              void* d_out, int out_size, void* d_ws, size_t ws_size,
                              hipStream_t stream) {
    (void)in_sizes; (void)n_in; (void)out_size; (void)ws_size;
    const float* f      = (const float*)d_in[0];
    const float* fneg   = (const float*)d_in[1];
    const int*   labels = (const int*)d_in[2];
    float*       out    = (float*)d_out;

    float* fnW   = (float*)d_ws;                       // 4096*128 floats
    float* gnW   = fnW + (size_t)B_ROWS * DIM;         // 8192*128 floats
    float* lossW = gnW + (size_t)M_ROWS * DIM;         // 4096 floats

    normalize_rows<<<(B_ROWS + M_ROWS) / WAVES_PER_BLOCK, 256, 0, stream>>>(
        f, fneg, fnW, gnW);
    supcon_kernel<<<B_ROWS / 16, 256, 0, stream>>>(fnW, gnW, labels, lossW);
    reduce_loss<<<1, 256, 0, stream>>>(lossW, out);
}